// LowRank_74835510165644
// MI455X (gfx1250) — compile-verified
//
#include <hip/hip_runtime.h>
#include <math.h>

// ||A - Pi_1(A)||_F / sqrt(m*n)  ==  sqrt(||A||_F^2 - lambda_max(A^T A)) / 512
// lambda_max via trace-normalized repeated squaring of the Gram matrix:
//   S1 = (G/tau0)^2 ; S_{j+1} = (S_j/tau_j)^2 ; tau_j = tr(S_j)
//   log(lambda1) = log(tau0) + sum_{j=1..K} 2^-j log(tau_j)
// All heavy work is 512x512x512 fp32 GEMMs -> v_wmma_f32_16x16x4_f32.

typedef __attribute__((ext_vector_type(2))) float v2f;
typedef __attribute__((ext_vector_type(8))) float v8f;

#define NDIM 512
#define KSQ  16   // number of squarings

// ---------------------------------------------------------------------------
// GEMM skeleton: 128x128 block tile, 256 threads (8 waves), K-tile = 32.
// Waves arranged 4 (M) x 2 (N); each wave owns 32x64 -> 2x4 WMMA accumulators.
// Per 4-wide k-step: 2 A-frag + 4 B-frag LDS loads feed 8 WMMAs.
// BOTH operand panels are stored K-contiguous in LDS (stride-33 padded rows),
// so every fragment is a contiguous float pair -> single ds_load_2addr_b32
// per operand, no repacking movs.
// ISA fragment layouts (CDNA5 ISA):
//   A 16x4 : lane L holds row M=L&15, K = vgpr + 2*(L>>4)
//   B 4x16 : lane L holds col N=L&15, K = vgpr + 2*(L>>4)
//   C/D    : lane L, vgpr r -> M = r + 8*(L>>4), N = L&15
// ---------------------------------------------------------------------------

#define LSTRIDE 33

// G = A^T * A   (per batch, blockIdx.z selects batch within chunk)
__global__ __launch_bounds__(256) void gram_kernel(const float* __restrict__ A,
                                                   float* __restrict__ G) {
  const float* Ab = A + (size_t)blockIdx.z * NDIM * NDIM;
  float*       Gb = G + (size_t)blockIdx.z * NDIM * NDIM;

  __shared__ float Xs[128 * LSTRIDE];  // Xs[m][k] = A[kt+k][m0b+m]
  __shared__ float Yt[128 * LSTRIDE];  // Yt[n][k] = A[kt+k][n0b+n]

  const int tid  = threadIdx.x;
  const int lane = tid & 31;
  const int wave = tid >> 5;
  const int l15  = lane & 15;
  const int hl   = lane >> 4;                 // lane half (0/1)
  const int m0w  = (wave >> 1) * 32;          // 4 waves along M
  const int n0w  = (wave & 1) * 64;           // 2 waves along N
  const int m0b  = blockIdx.y * 128;
  const int n0b  = blockIdx.x * 128;

  v8f acc[2][4] = {};

  for (int kt = 0; kt < NDIM; kt += 32) {
    __syncthreads();
    for (int idx = tid; idx < 128 * 32; idx += 256) {
      const int mn = idx & 127;
      const int k  = idx >> 7;
      const float* row = Ab + (size_t)(kt + k) * NDIM;
      Xs[mn * LSTRIDE + k] = row[m0b + mn];  // transposed store (coalesced read)
      Yt[mn * LSTRIDE + k] = row[n0b + mn];
    }
    __syncthreads();

#pragma unroll
    for (int kk = 0; kk < 32; kk += 4) {
      const int ko = kk + 2 * hl;
      v2f a[2], b[4];
#pragma unroll
      for (int mi = 0; mi < 2; ++mi) {
        const int row = m0w + mi * 16 + l15;
        a[mi].x = Xs[row * LSTRIDE + ko];
        a[mi].y = Xs[row * LSTRIDE + ko + 1];
      }
#pragma unroll
      for (int nj = 0; nj < 4; ++nj) {
        const int col = n0w + nj * 16 + l15;
        b[nj].x = Yt[col * LSTRIDE + ko];
        b[nj].y = Yt[col * LSTRIDE + ko + 1];
      }
#pragma unroll
      for (int mi = 0; mi < 2; ++mi)
#pragma unroll
        for (int nj = 0; nj < 4; ++nj)
          acc[mi][nj] = __builtin_amdgcn_wmma_f32_16x16x4_f32(
              false, a[mi], false, b[nj], (short)0, acc[mi][nj], false, false);
    }
  }

#pragma unroll
  for (int mi = 0; mi < 2; ++mi)
#pragma unroll
    for (int nj = 0; nj < 4; ++nj)
#pragma unroll
      for (int r = 0; r < 8; ++r) {
        const int gm = m0b + m0w + mi * 16 + r + 8 * hl;
        const int gn = n0b + n0w + nj * 16 + l15;
        Gb[(size_t)gm * NDIM + gn] = acc[mi][nj][r];
      }
}

// H = (G / tau_prev)^2  (scale folded into the accumulator epilogue)
__global__ __launch_bounds__(256) void square_kernel(const float* __restrict__ Gsrc,
                                                     float* __restrict__ Hdst,
                                                     const float* __restrict__ tau_prev) {
  const float* Gb = Gsrc + (size_t)blockIdx.z * NDIM * NDIM;
  float*       Hb = Hdst + (size_t)blockIdx.z * NDIM * NDIM;
  const float invt = 1.0f / tau_prev[blockIdx.z];
  const float scal = invt * invt;

  __shared__ float Xs[128 * LSTRIDE];  // Xs[m][k] = G[m0b+m][kt+k]
  __shared__ float Yt[128 * LSTRIDE];  // Yt[n][k] = G[kt+k][n0b+n]

  const int tid  = threadIdx.x;
  const int lane = tid & 31;
  const int wave = tid >> 5;
  const int l15  = lane & 15;
  const int hl   = lane >> 4;
  const int m0w  = (wave >> 1) * 32;
  const int n0w  = (wave & 1) * 64;
  const int m0b  = blockIdx.y * 128;
  const int n0b  = blockIdx.x * 128;

  v8f acc[2][4] = {};

  for (int kt = 0; kt < NDIM; kt += 32) {
    __syncthreads();
    for (int idx = tid; idx < 128 * 32; idx += 256) {
      const int k = idx & 31;
      const int m = idx >> 5;
      Xs[m * LSTRIDE + k] = Gb[(size_t)(m0b + m) * NDIM + kt + k];  // coalesced in k
    }
    for (int idx = tid; idx < 128 * 32; idx += 256) {
      const int n = idx & 127;
      const int k = idx >> 7;
      Yt[n * LSTRIDE + k] = Gb[(size_t)(kt + k) * NDIM + n0b + n];  // coalesced in n
    }
    __syncthreads();

#pragma unroll
    for (int kk = 0; kk < 32; kk += 4) {
      const int ko = kk + 2 * hl;
      v2f a[2], b[4];
#pragma unroll
      for (int mi = 0; mi < 2; ++mi) {
        const int row = m0w + mi * 16 + l15;
        a[mi].x = Xs[row * LSTRIDE + ko];
        a[mi].y = Xs[row * LSTRIDE + ko + 1];
      }
#pragma unroll
      for (int nj = 0; nj < 4; ++nj) {
        const int col = n0w + nj * 16 + l15;
        b[nj].x = Yt[col * LSTRIDE + ko];
        b[nj].y = Yt[col * LSTRIDE + ko + 1];
      }
#pragma unroll
      for (int mi = 0; mi < 2; ++mi)
#pragma unroll
        for (int nj = 0; nj < 4; ++nj)
          acc[mi][nj] = __builtin_amdgcn_wmma_f32_16x16x4_f32(
              false, a[mi], false, b[nj], (short)0, acc[mi][nj], false, false);
    }
  }

#pragma unroll
  for (int mi = 0; mi < 2; ++mi)
#pragma unroll
    for (int nj = 0; nj < 4; ++nj)
#pragma unroll
      for (int r = 0; r < 8; ++r) {
        const int gm = m0b + m0w + mi * 16 + r + 8 * hl;
        const int gn = n0b + n0w + nj * 16 + l15;
        Hb[(size_t)gm * NDIM + gn] = acc[mi][nj][r] * scal;
      }
}

// tau[b] = trace of 512x512 matrix b
__global__ __launch_bounds__(256) void trace_kernel(const float* __restrict__ M,
                                                    float* __restrict__ tau) {
  __shared__ float red[256];
  const int tid = threadIdx.x;
  const float* Mb = M + (size_t)blockIdx.x * NDIM * NDIM;
  float s = Mb[(size_t)tid * (NDIM + 1)] + Mb[(size_t)(tid + 256) * (NDIM + 1)];
  red[tid] = s;
  __syncthreads();
  for (int ofs = 128; ofs > 0; ofs >>= 1) {
    if (tid < ofs) red[tid] += red[tid + ofs];
    __syncthreads();
  }
  if (tid == 0) tau[blockIdx.x] = red[0];
}

// out[b] = sqrt(tau0 - exp(log tau0 + sum 2^-j log tau_j)) / 512
__global__ __launch_bounds__(64) void finalize_kernel(const float* __restrict__ tau,
                                                      float* __restrict__ out,
                                                      int cnt, int chstride) {
  const int t = threadIdx.x;
  if (t >= cnt) return;
  const float t0 = tau[t];
  float lacc = logf(t0);
  float w = 1.0f;
  for (int j = 1; j <= KSQ; ++j) {
    w *= 0.5f;
    lacc += w * logf(tau[j * chstride + t]);
  }
  const float lam = expf(lacc);
  out[t] = sqrtf(fmaxf(t0 - lam, 0.0f)) * (1.0f / 512.0f);
}

extern "C" void kernel_launch(void* const* d_in, const int* in_sizes, int n_in,
                              void* d_out, int out_size, void* d_ws, size_t ws_size,
                              hipStream_t stream) {
  const float* x = (const float*)d_in[0];
  float* out = (float*)d_out;
  float* w = (float*)d_ws;

  const size_t MAT = (size_t)NDIM * NDIM;           // 262144 floats per matrix
  const int nbatch = in_sizes[0] / (int)MAT;        // 128

  // Chunk size: need 2 ping-pong G buffers + (KSQ+1)*ch trace slots in ws.
  int ch = (int)(ws_size / (sizeof(float) * (2 * MAT + (KSQ + 1))));
  if (ch > 32) ch = 32;
  if (ch > nbatch) ch = nbatch;
  if (ch < 1) ch = 1;

  float* G0  = w;
  float* G1  = w + (size_t)ch * MAT;
  float* tau = w + 2 * (size_t)ch * MAT;            // layout: tau[j*ch + b]

  for (int c = 0; c < nbatch; c += ch) {
    const int cnt = (nbatch - c < ch) ? (nbatch - c) : ch;
    const float* Ac = x + (size_t)c * MAT;

    dim3 grid(NDIM / 128, NDIM / 128, cnt);
    dim3 blk(256);

    gram_kernel<<<grid, blk, 0, stream>>>(Ac, G0);
    trace_kernel<<<dim3(cnt), dim3(256), 0, stream>>>(G0, tau);

    float* src = G0;
    float* dst = G1;
    for (int j = 1; j <= KSQ; ++j) {
      square_kernel<<<grid, blk, 0, stream>>>(src, dst, tau + (size_t)(j - 1) * ch);
      trace_kernel<<<dim3(cnt), dim3(256), 0, stream>>>(dst, tau + (size_t)j * ch);
      float* tmp = src; src = dst; dst = tmp;
    }

    finalize_kernel<<<1, 64, 0, stream>>>(tau, out + c, cnt, ch);
  }
}